// CLIP_3v3d_brats_55825984913444
// MI455X (gfx1250) — compile-verified
//
#include <hip/hip_runtime.h>
#include <math.h>

typedef __attribute__((ext_vector_type(2))) float v2f;
typedef __attribute__((ext_vector_type(8))) float v8f;

#define S3   884736   // 96^3
#define Q3   221184   // 96^3 / 4 (float4 quads)
#define CVOL 13824    // 24^3
#define GVOL 110592   // 8 * 24^3 (one GN group, contiguous)

// workspace layout (float offsets)
#define WS_MU    0        // [32]   per (b,group) mean
#define WS_RSTD  32       // [32]   per (b,group) rsqrt(var+eps)
#define WS_S     64       // [2][128][27] patch sums
#define WS_XF    6976     // [2][256] x_feat
#define WS_EFFW  7488     // [6][32] eff_w rows = b*3+k
// total 7680 floats = 30 KB

// ---------------------------------------------------------------------------
// Kernel 1: GroupNorm statistics. grid = 32 (b*16+g), block = 256.
// Each group = 8 contiguous channels = 110592 contiguous floats.
// ---------------------------------------------------------------------------
__global__ void __launch_bounds__(256) k_gn_stats(const float* __restrict__ xe,
                                                  float* __restrict__ ws) {
    int bg = blockIdx.x;                      // b*16 + g
    const float4* p = (const float4*)(xe + (size_t)bg * GVOL);
    float s = 0.f, ss = 0.f;
    for (int i = threadIdx.x; i < GVOL / 4; i += 256) {
        float4 v = p[i];
        s  += v.x + v.y + v.z + v.w;
        ss += v.x * v.x + v.y * v.y + v.z * v.z + v.w * v.w;
    }
    for (int off = 16; off > 0; off >>= 1) {
        s  += __shfl_down(s, off, 32);
        ss += __shfl_down(ss, off, 32);
    }
    __shared__ float rs[8], rss[8];
    int wave = threadIdx.x >> 5, lane = threadIdx.x & 31;
    if (lane == 0) { rs[wave] = s; rss[wave] = ss; }
    __syncthreads();
    if (threadIdx.x == 0) {
        float S = 0.f, SS = 0.f;
        for (int w = 0; w < 8; ++w) { S += rs[w]; SS += rss[w]; }
        float mu  = S / (float)GVOL;
        float var = SS / (float)GVOL - mu * mu;
        ws[WS_MU + bg]   = mu;
        ws[WS_RSTD + bg] = rsqrtf(var + 1e-5f);
    }
}

// ---------------------------------------------------------------------------
// Kernel 2: per-(b,channel) GN+ReLU into LDS, then the 27 strided patch sums
// S[b,ci,kd,kw,kh] = sum over 11^3 output positions of the 2-strided input.
// grid = 256 (b*128+ci), block = 256. LDS = 55 KB.
// ---------------------------------------------------------------------------
__global__ void __launch_bounds__(256) k_patchsum(const float* __restrict__ xe,
                                                  const float* __restrict__ gn_g,
                                                  const float* __restrict__ gn_b,
                                                  float* __restrict__ ws) {
    int bc = blockIdx.x;                      // b*128 + ci
    int b  = bc >> 7, ci = bc & 127;
    int g  = ci >> 3;
    float mu   = ws[WS_MU + b * 16 + g];
    float rstd = ws[WS_RSTD + b * 16 + g];
    float ga = gn_g[ci], be = gn_b[ci];

    __shared__ __align__(16) float v[CVOL];
    __shared__ float wred[27][8];

    const float4* src = (const float4*)(xe + (size_t)bc * CVOL);
    for (int i = threadIdx.x; i < CVOL / 4; i += 256) {
        float4 x = src[i];
        float4 y;
        y.x = fmaxf((x.x - mu) * rstd * ga + be, 0.f);
        y.y = fmaxf((x.y - mu) * rstd * ga + be, 0.f);
        y.z = fmaxf((x.z - mu) * rstd * ga + be, 0.f);
        y.w = fmaxf((x.w - mu) * rstd * ga + be, 0.f);
        ((float4*)v)[i] = y;
    }
    __syncthreads();

    float acc[27];
#pragma unroll
    for (int j = 0; j < 27; ++j) acc[j] = 0.f;

    for (int p = threadIdx.x; p < 1331; p += 256) {
        int sd = p / 121;
        int r  = p - sd * 121;
        int sw = r / 11;
        int sh = r - sw * 11;
        const float* base = v + (2 * sd) * 576 + (2 * sw) * 24 + 2 * sh;
#pragma unroll
        for (int kd = 0; kd < 3; ++kd)
#pragma unroll
            for (int kw = 0; kw < 3; ++kw)
#pragma unroll
                for (int kh = 0; kh < 3; ++kh)
                    acc[kd * 9 + kw * 3 + kh] += base[kd * 576 + kw * 24 + kh];
    }

    int wave = threadIdx.x >> 5, lane = threadIdx.x & 31;
#pragma unroll
    for (int j = 0; j < 27; ++j) {
        float a = acc[j];
        for (int off = 16; off > 0; off >>= 1) a += __shfl_down(a, off, 32);
        if (lane == 0) wred[j][wave] = a;
    }
    __syncthreads();
    if (threadIdx.x < 27) {
        float t = 0.f;
#pragma unroll
        for (int w = 0; w < 8; ++w) t += wred[threadIdx.x][w];
        ws[WS_S + bc * 27 + threadIdx.x] = t;
    }
}

// ---------------------------------------------------------------------------
// Kernel 3: x_feat[b,o] = gap_b[o] + (1/1331) * dot(S[b,:], gap_w[o,:]).
// grid = 512 (b*256+o), block = 128.
// ---------------------------------------------------------------------------
__global__ void __launch_bounds__(128) k_xfeat(const float* __restrict__ gap_w,
                                               const float* __restrict__ gap_b,
                                               float* __restrict__ ws) {
    int bo = blockIdx.x;                       // b*256 + o
    int b = bo >> 8, o = bo & 255;
    const float* Sb = ws + WS_S + b * 3456;
    const float* W  = gap_w + (size_t)o * 3456;
    float acc = 0.f;
    for (int j = threadIdx.x; j < 3456; j += 128) acc += Sb[j] * W[j];
    for (int off = 16; off > 0; off >>= 1) acc += __shfl_down(acc, off, 32);
    __shared__ float r[4];
    int wave = threadIdx.x >> 5, lane = threadIdx.x & 31;
    if (lane == 0) r[wave] = acc;
    __syncthreads();
    if (threadIdx.x == 0)
        ws[WS_XF + bo] = gap_b[o] + (r[0] + r[1] + r[2] + r[3]) * (1.0f / 1331.0f);
}

// ---------------------------------------------------------------------------
// Kernel 4: fused controller head via V_WMMA_F32_16X16X4_F32.
// One workgroup, 8 waves. M=16 (rows 0..5 = (b,k), rest zero-padded).
//   P[16,512] = relu(X @ Wcf^T + b_cf)
//   C[16,256] =      P @ Wc^T  + b_c
//   H[16,16]  = relu(C @ Wa1^T + b_a1)
//   G[16,32]  = sigm(H @ Wa2^T + b_a2);  eff_w = G * w_seg
// Fragment layout per ISA 7.12.2:
//   A 16x4 f32:  M = lane&15, vgpr v holds K = 2*(lane>>4)+v
//   B 4x16 f32:  N = lane&15, vgpr v holds K = 2*(lane>>4)+v
//   C/D 16x16:   N = lane&15, vgpr r holds M = r + 8*(lane>>4)
// Padding rows are handled by pointing their lanes at a zeroed LDS row so all
// fragment loads are unconditional ds_load_b64 (no EXEC-predicated loads).
// ---------------------------------------------------------------------------
#define PLD 514
#define CLD 258
#define HLD 18

__global__ void __launch_bounds__(256) k_head(const float* __restrict__ w_cf,
                                              const float* __restrict__ b_cf,
                                              const float* __restrict__ w_c,
                                              const float* __restrict__ b_c,
                                              const float* __restrict__ w_a1,
                                              const float* __restrict__ b_a1,
                                              const float* __restrict__ w_a2,
                                              const float* __restrict__ b_a2,
                                              const float* __restrict__ emb,
                                              const float* __restrict__ w_seg,
                                              float* __restrict__ ws) {
    __shared__ __align__(16) float xfL[512];   // x_feat [2][256]
    __shared__ __align__(16) float embL[768];  // emb    [3][256]
    __shared__ __align__(16) float zrow[256];  // zero row for padding lanes
    __shared__ __align__(16) float Ps[16 * PLD];
    __shared__ __align__(16) float Cs[16 * CLD];
    __shared__ __align__(16) float Hs[16 * HLD];

    int tid = threadIdx.x;
    for (int i = tid; i < 512; i += 256) xfL[i] = ws[WS_XF + i];
    for (int i = tid; i < 768; i += 256) embL[i] = emb[i];
    if (tid < 256) zrow[tid] = 0.f;
    __syncthreads();

    int wave = tid >> 5, lane = tid & 31;
    int half = lane >> 4, l16 = lane & 15;

    // per-lane A row pointers for stage 1 (row = l16 = b*3+k; rows >=6 zero)
    bool rowvalid = (l16 < 6);
    int rowb = l16 / 3, rowk = l16 - rowb * 3;
    const float* arow0 = rowvalid ? (xfL + rowb * 256) : zrow;  // K < 256
    const float* arow1 = rowvalid ? (embL + rowk * 256) : zrow; // K >= 256

    // ---- Stage 1: P = relu(X @ Wcf^T + b_cf), 32 N-tiles / 8 waves ----
    for (int nt = wave; nt < 32; nt += 8) {
        int nb = nt * 16;
        const float* wrow = w_cf + (size_t)(nb + l16) * 512;
        v8f acc = {};
        for (int ks = 0; ks < 64; ++ks) {               // K = 0..255  (x_feat)
            int kb = ks * 4 + 2 * half;
            v2f a  = *(const v2f*)(arow0 + kb);
            v2f bb = *(const v2f*)(wrow + kb);
            acc = __builtin_amdgcn_wmma_f32_16x16x4_f32(false, a, false, bb,
                                                        (short)0, acc, false, false);
        }
        for (int ks = 0; ks < 64; ++ks) {               // K = 256..511 (emb)
            int kb = ks * 4 + 2 * half;
            v2f a  = *(const v2f*)(arow1 + kb);
            v2f bb = *(const v2f*)(wrow + 256 + kb);
            acc = __builtin_amdgcn_wmma_f32_16x16x4_f32(false, a, false, bb,
                                                        (short)0, acc, false, false);
        }
        float bias = b_cf[nb + l16];
#pragma unroll
        for (int r = 0; r < 8; ++r) {
            int m = r + 8 * half;
            Ps[m * PLD + nb + l16] = fmaxf(acc[r] + bias, 0.f);
        }
    }
    __syncthreads();

    // ---- Stage 2: C = P @ Wc^T + b_c, 16 N-tiles / 8 waves ----
    for (int nt = wave; nt < 16; nt += 8) {
        int nb = nt * 16;
        const float* wrow = w_c + (size_t)(nb + l16) * 512;
        const float* arow = Ps + l16 * PLD;
        v8f acc = {};
        for (int ks = 0; ks < 128; ++ks) {
            int kb = ks * 4 + 2 * half;
            v2f a  = *(const v2f*)(arow + kb);
            v2f bb = *(const v2f*)(wrow + kb);
            acc = __builtin_amdgcn_wmma_f32_16x16x4_f32(false, a, false, bb,
                                                        (short)0, acc, false, false);
        }
        float bias = b_c[nb + l16];
#pragma unroll
        for (int r = 0; r < 8; ++r) {
            int m = r + 8 * half;
            Cs[m * CLD + nb + l16] = acc[r] + bias;
        }
    }
    __syncthreads();

    // ---- Stage 3: H = relu(C @ Wa1^T + b_a1), 1 tile, wave 0 ----
    if (wave == 0) {
        const float* wrow = w_a1 + (size_t)l16 * 256;
        const float* arow = Cs + l16 * CLD;
        v8f acc = {};
        for (int ks = 0; ks < 64; ++ks) {
            int kb = ks * 4 + 2 * half;
            v2f a  = *(const v2f*)(arow + kb);
            v2f bb = *(const v2f*)(wrow + kb);
            acc = __builtin_amdgcn_wmma_f32_16x16x4_f32(false, a, false, bb,
                                                        (short)0, acc, false, false);
        }
        float bias = b_a1[l16];
#pragma unroll
        for (int r = 0; r < 8; ++r) {
            int m = r + 8 * half;
            Hs[m * HLD + l16] = fmaxf(acc[r] + bias, 0.f);
        }
    }
    __syncthreads();

    // ---- Stage 4: gate = sigmoid(H @ Wa2^T + b_a2); eff_w = gate*w_seg ----
    if (wave < 2) {
        int nb = wave * 16;
        const float* wrow = w_a2 + (size_t)(nb + l16) * 16;
        const float* arow = Hs + l16 * HLD;
        v8f acc = {};
#pragma unroll
        for (int ks = 0; ks < 4; ++ks) {
            int kb = ks * 4 + 2 * half;
            v2f a  = *(const v2f*)(arow + kb);
            v2f bb = *(const v2f*)(wrow + kb);
            acc = __builtin_amdgcn_wmma_f32_16x16x4_f32(false, a, false, bb,
                                                        (short)0, acc, false, false);
        }
        float bias = b_a2[nb + l16];
#pragma unroll
        for (int r = 0; r < 8; ++r) {
            int m = r + 8 * half;
            if (m < 6) {
                int k = m % 3;
                float gate = 1.0f / (1.0f + __expf(-(acc[r] + bias)));
                ws[WS_EFFW + m * 32 + nb + l16] = gate * w_seg[k * 32 + nb + l16];
            }
        }
    }
}

// ---------------------------------------------------------------------------
// Kernel 5: logits[b,k,s] = sum_c eff_w[b,k,c]*pred[b,c,s] + b_seg[k].
// Pure HBM stream: 226 MB read + 21 MB write ~ 10.6 us floor at 23.3 TB/s.
// grid = 1728 (864 per batch), block = 256, one float4 quad per thread.
// ---------------------------------------------------------------------------
__global__ void __launch_bounds__(256) k_logits(const float* __restrict__ pred,
                                                const float* __restrict__ b_seg,
                                                const float* __restrict__ ws,
                                                float* __restrict__ out) {
    int blk = blockIdx.x;
    int b = blk / 864;
    int q = (blk - b * 864) * 256 + threadIdx.x;   // quad index in 96^3/4

    __shared__ float ew[96];                        // [k][c]
    if (threadIdx.x < 96) ew[threadIdx.x] = ws[WS_EFFW + b * 96 + threadIdx.x];
    __syncthreads();

    const float4* pv = (const float4*)pred + (size_t)b * 32 * Q3 + q;
    float bs0 = b_seg[0], bs1 = b_seg[1], bs2 = b_seg[2];
    float4 a0 = make_float4(bs0, bs0, bs0, bs0);
    float4 a1 = make_float4(bs1, bs1, bs1, bs1);
    float4 a2 = make_float4(bs2, bs2, bs2, bs2);

#pragma unroll
    for (int c = 0; c < 32; ++c) {
        if (c + 8 < 32)
            __builtin_prefetch((const void*)(pv + (size_t)(c + 8) * Q3), 0, 0);
        float4 p = pv[(size_t)c * Q3];
        float w0 = ew[c], w1 = ew[32 + c], w2 = ew[64 + c];
        a0.x += w0 * p.x; a0.y += w0 * p.y; a0.z += w0 * p.z; a0.w += w0 * p.w;
        a1.x += w1 * p.x; a1.y += w1 * p.y; a1.z += w1 * p.z; a1.w += w1 * p.w;
        a2.x += w2 * p.x; a2.y += w2 * p.y; a2.z += w2 * p.z; a2.w += w2 * p.w;
    }

    float4* ov = (float4*)out + (size_t)b * 3 * Q3 + q;
    ov[0]                = a0;
    ov[(size_t)Q3]       = a1;
    ov[(size_t)2 * Q3]   = a2;
}

// ---------------------------------------------------------------------------
extern "C" void kernel_launch(void* const* d_in, const int* in_sizes, int n_in,
                              void* d_out, int out_size, void* d_ws, size_t ws_size,
                              hipStream_t stream) {
    const float* x_e   = (const float*)d_in[0];
    const float* pred  = (const float*)d_in[1];
    const float* gn_g  = (const float*)d_in[2];
    const float* gn_b  = (const float*)d_in[3];
    const float* gap_w = (const float*)d_in[4];
    const float* gap_b = (const float*)d_in[5];
    const float* w_cf  = (const float*)d_in[6];
    const float* b_cf  = (const float*)d_in[7];
    const float* w_c   = (const float*)d_in[8];
    const float* b_c   = (const float*)d_in[9];
    const float* w_a1  = (const float*)d_in[10];
    const float* b_a1  = (const float*)d_in[11];
    const float* w_a2  = (const float*)d_in[12];
    const float* b_a2  = (const float*)d_in[13];
    const float* emb   = (const float*)d_in[14];
    const float* w_seg = (const float*)d_in[15];
    const float* b_seg = (const float*)d_in[16];
    float* out = (float*)d_out;
    float* ws  = (float*)d_ws;   // needs 30 KB

    k_gn_stats<<<32, 256, 0, stream>>>(x_e, ws);
    k_patchsum<<<256, 256, 0, stream>>>(x_e, gn_g, gn_b, ws);
    k_xfeat<<<512, 128, 0, stream>>>(gap_w, gap_b, ws);
    k_head<<<1, 256, 0, stream>>>(w_cf, b_cf, w_c, b_c, w_a1, b_a1,
                                  w_a2, b_a2, emb, w_seg, ws);
    k_logits<<<1728, 256, 0, stream>>>(pred, b_seg, ws, out);
}